// GeoEEGSNN_43937515438184
// MI455X (gfx1250) — compile-verified
//
#include <hip/hip_runtime.h>

typedef __attribute__((ext_vector_type(2))) float v2f;
typedef __attribute__((ext_vector_type(8))) float v8f;

#define B_  32
#define C_  64
#define T_  4000
#define TP_ 1000
#define BC_ (B_*C_)

#define INV_SQRT_1PEPS 0.9999950000374997f  // 1/sqrt(1+1e-5)

__device__ __forceinline__ float elu1(float x) {
    return x > 0.f ? x : (__expf(x) - 1.f);
}

// ---------------------------------------------------------------------------
// K1: gaussian encode + fusion + bn2 + elu  ->  S (B,C,T)
// ---------------------------------------------------------------------------
__global__ void k_encode(const float* __restrict__ xb, const float* __restrict__ mu,
                         const float* __restrict__ sigma, const float* __restrict__ fw,
                         const float* __restrict__ bn2_g, const float* __restrict__ bn2_b,
                         float* __restrict__ S)
{
    int idx = blockIdx.x * blockDim.x + threadIdx.x;   // b*C*T + c*T + t
    if (idx >= BC_ * T_) return;
    int t = idx % T_;
    int c = (idx / T_) % C_;
    int b = idx / (T_ * C_);

    float muv[4], is2[4];
#pragma unroll
    for (int n = 0; n < 4; n++) {
        muv[n] = mu[n];
        float sg = sigma[n];
        is2[n] = 1.f / (2.f * sg * sg + 1e-6f);
    }
    float s = 0.f;
#pragma unroll
    for (int ch = 0; ch < 3; ch++) {
        float x = xb[((b * 3 + ch) * C_ + c) * T_ + t];
#pragma unroll
        for (int n = 0; n < 4; n++) {
            float d = x - muv[n];
            s += fw[ch * 4 + n] * __expf(-d * d * is2[n]);
        }
    }
    s = s * INV_SQRT_1PEPS * bn2_g[0] + bn2_b[0];
    S[idx] = elu1(s);
}

// ---------------------------------------------------------------------------
// K1b: avg-pool-4 of curvature/tangent raw channels -> (tp, b, c) layout
// ---------------------------------------------------------------------------
__global__ void k_pool_raw(const float* __restrict__ xb,
                           float* __restrict__ curv, float* __restrict__ tang)
{
    int idx = blockIdx.x * blockDim.x + threadIdx.x;   // tp*BC + b*C + c
    if (idx >= BC_ * TP_) return;
    int c  = idx % C_;
    int b  = (idx / C_) % B_;
    int tp = idx / (C_ * B_);
    const float* pc = xb + ((b * 3 + 1) * C_ + c) * T_ + 4 * tp;
    const float* pt = xb + ((b * 3 + 2) * C_ + c) * T_ + 4 * tp;
    float sc = 0.f, st = 0.f;
#pragma unroll
    for (int j = 0; j < 4; j++) { sc += pc[j]; st += pt[j]; }
    curv[idx] = sc * 0.25f;
    tang[idx] = st * 0.25f;
}

// ---------------------------------------------------------------------------
// K2: depthwise conv k=16, pad 8 (cross-correlation, XLA convention), t' < 4000
// ---------------------------------------------------------------------------
__global__ void k_dwconv(const float* __restrict__ S, const float* __restrict__ dw,
                         float* __restrict__ Y0)
{
    int idx = blockIdx.x * blockDim.x + threadIdx.x;   // b*C*T + c*T + t'
    if (idx >= BC_ * T_) return;
    int t = idx % T_;
    int c = (idx / T_) % C_;
    const float* sp = S + (idx - t);                   // row (b,c)
    const float* w  = dw + c * 16;
    float acc = 0.f;
#pragma unroll
    for (int k = 0; k < 16; k++) {
        int ts = t + k - 8;
        if (ts >= 0 && ts < T_) acc += w[k] * sp[ts];
    }
    Y0[idx] = acc;
}

// ---------------------------------------------------------------------------
// K3: pointwise matmul via V_WMMA_F32_16X16X4_F32, fused bn1 + elu.
// Block = 128 threads (4 waves) handles one (b, 16-wide t tile); wave w owns
// output rows d0 = w*16. Only rows 0-15 of the 64x16 B panel (Y0[b, :, t0:t0+16]) are
// staged into LDS ONCE per block via the CDNA5 async DMA path
// (GLOBAL_LOAD_ASYNC_TO_LDS_B128, ASYNCcnt-tracked). The LDS offset operand is
// derived from the REAL &Ys[...] pointer (ptrtoint -> low 32 bits are the LDS
// byte offset per the aperture scheme) so the shared array escapes and the
// "memory" clobber keeps the consumer ds_loads alive.
// Two independent accumulator chains (even/odd k) double the WMMA dependency
// distance, eliminating the D->C RAW v_nop stalls seen with a single chain.
//   A (16x4, MxK): lanes 0-15 = rows M, VGPR0/1 = K0/K1; lanes 16-31 = K2/K3.
//   B (4x16, KxN): mirrored (lane = column N, half selects K pair).
//   C/D: VGPR r -> M = r (+8 for upper half), N = lane&15.
// ---------------------------------------------------------------------------
__global__ void __launch_bounds__(128)
k_pwmm_wmma(const float* __restrict__ Y0, const float* __restrict__ W,
            const float* __restrict__ bn1_g, const float* __restrict__ bn1_b,
            float* __restrict__ Yact)
{
    __shared__ float Ys[C_ * 16];       // Ys[c*16 + n] = Y0[b][c][t0+n]

    int tile = blockIdx.x;              // B_ * 250 tiles
    int tt   = tile % 250;
    int b    = tile / 250;
    int t0   = tt * 16;

    int tid  = threadIdx.x;
    int wave = tid >> 5;
    int lane = tid & 31;
    int half = lane >> 4;               // 0: K0/K1, 1: K2/K3
    int lr   = lane & 15;               // M row (A) / N column (B)
    int d0   = wave * 16;

    const float* Yb = Y0 + b * (C_ * T_);

    // ---- stage B panel: 16 rows x 16 floats, one b128 DMA per row-quarter ----
    if (tid < 64) {
        int c   = tid >> 2;             // row (input channel)
        int seg = (tid & 3) * 4;        // 4-float segment within the row
        const float* g = Yb + c * T_ + t0 + seg;
        // low 32 bits of the generic shared pointer == LDS byte offset;
        // the ptrtoint escapes Ys so alias analysis honors the memory clobber
        unsigned ldsoff = (unsigned)(unsigned long long)(&Ys[c * 16 + seg]);
        asm volatile("global_load_async_to_lds_b128 %0, %1, off"
                     :: "v"(ldsoff), "v"(g)
                     : "memory");
    }
    asm volatile("s_wait_asynccnt 0x0" ::: "memory");
    __syncthreads();

    v8f acc0 = {}, acc1 = {};
#pragma unroll
    for (int k = 0; k < 16; k += 2) {
        int kb0 = k * 4 + half * 2;
        int kb1 = (k + 1) * 4 + half * 2;
        v2f a0, b0, a1, b1;
        a0.x = W[(d0 + lr) * C_ + kb0];
        a0.y = W[(d0 + lr) * C_ + kb0 + 1];
        b0.x = Ys[kb0 * 16 + lr];
        b0.y = Ys[(kb0 + 1) * 16 + lr];
        a1.x = W[(d0 + lr) * C_ + kb1];
        a1.y = W[(d0 + lr) * C_ + kb1 + 1];
        b1.x = Ys[kb1 * 16 + lr];
        b1.y = Ys[(kb1 + 1) * 16 + lr];
        acc0 = __builtin_amdgcn_wmma_f32_16x16x4_f32(
                   false, a0, false, b0, (short)0, acc0, false, false);
        acc1 = __builtin_amdgcn_wmma_f32_16x16x4_f32(
                   false, a1, false, b1, (short)0, acc1, false, false);
    }

#pragma unroll
    for (int r = 0; r < 8; r++) {
        int d   = d0 + r + half * 8;
        float v = (acc0[r] + acc1[r]) * (bn1_g[d] * INV_SQRT_1PEPS) + bn1_b[d];
        v = elu1(v);
        Yact[(b * C_ + d) * T_ + t0 + lr] = v;
    }
}

// ---------------------------------------------------------------------------
// K4: avg-pool-4 of activated y -> CB in (tp, b, c) layout
// ---------------------------------------------------------------------------
__global__ void k_pool_y(const float* __restrict__ Y, float* __restrict__ CB)
{
    int idx = blockIdx.x * blockDim.x + threadIdx.x;
    if (idx >= BC_ * TP_) return;
    int tp = idx % TP_;
    int c  = (idx / TP_) % C_;
    int b  = idx / (TP_ * C_);
    const float* p = Y + (b * C_ + c) * T_ + 4 * tp;
    CB[tp * BC_ + b * C_ + c] = 0.25f * (p[0] + p[1] + p[2] + p[3]);
}

// ---------------------------------------------------------------------------
// K5: softmax-attn stats + LIF scan + attention-weighted spike sum
// one thread per (b,c) chain; all buffers (tp,b,c) so lanes are coalesced in c
// ---------------------------------------------------------------------------
__global__ void k_scan(const float* __restrict__ CURV, const float* __restrict__ TANG,
                       const float* __restrict__ CB,
                       const float* __restrict__ alpha_p, const float* __restrict__ gamma_p,
                       const float* __restrict__ beta0_p, float* __restrict__ FE)
{
    int i = blockIdx.x * blockDim.x + threadIdx.x;
    if (i >= BC_) return;
    float alpha = alpha_p[0], gamma = gamma_p[0], beta0 = beta0_p[0];

    float mx = -3.402823466e+38f;
    for (int t = 0; t < TP_; t++) mx = fmaxf(mx, CURV[t * BC_ + i]);
    float se = 0.f;
    for (int t = 0; t < TP_; t++) se += __expf(CURV[t * BC_ + i] - mx);
    float inv_se = 1.f / se;

    float m = 0.f, feat = 0.f;
    for (int t = 0; t < TP_; t++) {
        float cu = CURV[t * BC_ + i];
        float ta = TANG[t * BC_ + i];
        float tc = fminf(fmaxf(ta, -2.f), 2.f);
        float cc = fminf(fmaxf(cu, -2.f), 2.f);
        float vth  = fmaxf(1.f + alpha * tc, 0.1f);
        float beta = 1.f / (1.f + __expf(-(beta0 - gamma * cc)));
        m = beta * m + CB[t * BC_ + i];
        float spk = (m > vth) ? 1.f : 0.f;
        m -= spk * vth;
        feat += spk * __expf(cu - mx) * inv_se;
    }
    FE[i] = feat;
}

// ---------------------------------------------------------------------------
// K6: final FC (B,C) x (4,C)^T + bias -> out (B,4)
// ---------------------------------------------------------------------------
__global__ void k_fc(const float* __restrict__ FE, const float* __restrict__ fcw,
                     const float* __restrict__ fcb, float* __restrict__ out)
{
    int idx = threadIdx.x;
    if (idx >= B_ * 4) return;
    int j = idx % 4;
    int b = idx / 4;
    float s = fcb[j];
    for (int c = 0; c < C_; c++) s += FE[b * C_ + c] * fcw[j * C_ + c];
    out[b * 4 + j] = s;
}

// ---------------------------------------------------------------------------
extern "C" void kernel_launch(void* const* d_in, const int* in_sizes, int n_in,
                              void* d_out, int out_size, void* d_ws, size_t ws_size,
                              hipStream_t stream)
{
    const float* xb    = (const float*)d_in[0];
    const float* mu    = (const float*)d_in[1];
    const float* sigma = (const float*)d_in[2];
    const float* fw    = (const float*)d_in[3];
    const float* bn2g  = (const float*)d_in[4];
    const float* bn2b  = (const float*)d_in[5];
    const float* dww   = (const float*)d_in[6];
    const float* pww   = (const float*)d_in[7];
    const float* bn1g  = (const float*)d_in[8];
    const float* bn1b  = (const float*)d_in[9];
    const float* alpha = (const float*)d_in[10];
    const float* gamma = (const float*)d_in[11];
    const float* beta0 = (const float*)d_in[12];
    const float* fcw   = (const float*)d_in[13];
    const float* fcb   = (const float*)d_in[14];

    float* ws   = (float*)d_ws;
    float* S    = ws;                       // B*C*T   (later reused as Yact)
    float* Y0   = S    + (size_t)BC_ * T_;  // B*C*T
    float* CURV = Y0   + (size_t)BC_ * T_;  // TP*B*C
    float* TANG = CURV + (size_t)BC_ * TP_; // TP*B*C
    float* CB   = TANG + (size_t)BC_ * TP_; // TP*B*C
    float* FE   = CB   + (size_t)BC_ * TP_; // B*C

    const int nBCT = BC_ * T_;      // 8,192,000
    const int nPBC = BC_ * TP_;     // 2,048,000

    k_encode  <<<(nBCT + 255) / 256, 256, 0, stream>>>(xb, mu, sigma, fw, bn2g, bn2b, S);
    k_pool_raw<<<(nPBC + 255) / 256, 256, 0, stream>>>(xb, CURV, TANG);
    k_dwconv  <<<(nBCT + 255) / 256, 256, 0, stream>>>(S, dww, Y0);
    k_pwmm_wmma<<<B_ * 250, 128, 0, stream>>>(Y0, pww, bn1g, bn1b, S);  // Yact -> S
    k_pool_y  <<<(nPBC + 255) / 256, 256, 0, stream>>>(S, CB);
    k_scan    <<<(BC_ + 255) / 256, 256, 0, stream>>>(CURV, TANG, CB, alpha, gamma, beta0, FE);
    k_fc      <<<1, 128, 0, stream>>>(FE, fcw, fcb, (float*)d_out);
}